// HungarianMatcher_58789512347998
// MI455X (gfx1250) — compile-verified
//
#include <hip/hip_runtime.h>
#include <hip/hip_bf16.h>
#include <stdint.h>

// Problem dims (match reference)
#define BSd   32
#define Qd    600
#define NCd   152
#define Td    32
#define NTd   1024                 // BSd*Td
#define ROWS  (BSd * Qd)           // 19200
#define CELEM ((size_t)ROWS * NTd) // 19660800 per cost matrix
#define TILE_R 16
#define NTILES (ROWS / TILE_R)     // 1200

typedef float    v2f __attribute__((ext_vector_type(2)));
typedef float    v8f __attribute__((ext_vector_type(8)));
typedef uint32_t v4u __attribute__((ext_vector_type(4)));
typedef int32_t  v8i __attribute__((ext_vector_type(8)));
typedef int32_t  v4i __attribute__((ext_vector_type(4)));

// ---------------------------------------------------------------------------
// TDM: issue a 2D tensor tile load (global -> LDS) via the Tensor Data Mover.
// D# packed per CDNA5 ISA ch.8: group0 = {flags, lds_addr, global_addr, type=2},
// group1 = {data_size/pad, tensor dims, tile dims, strides}, groups 2/3 = 0 (2D).
// This toolchain exposes the 6-arg builtin (extra trailing int32x8 group, which
// we zero — only needed for >2D tensors). data_size = 4 bytes (code 2).
// Caller must be wave-uniform; wave 0 only.
// ---------------------------------------------------------------------------
__device__ __forceinline__ void tdm_load_2d(uint32_t lds_addr, uint64_t gaddr,
                                            uint32_t tensor_d0, uint32_t tensor_d1,
                                            uint32_t tile_d0, uint32_t tile_d1,
                                            uint64_t stride0_elems,
                                            uint32_t pad_en, uint32_t pad_interval,
                                            uint32_t pad_amount)
{
    v4u g0; v8i g1;
    v4i g2 = {0, 0, 0, 0};
    v4i g3 = {0, 0, 0, 0};
    v8i g4 = {0, 0, 0, 0, 0, 0, 0, 0};
    g0[0] = 1u;                                                   // count=1, user mode
    g0[1] = lds_addr;                                             // lds_addr[31:0]
    g0[2] = (uint32_t)gaddr;                                      // global_addr[31:0]
    g0[3] = ((uint32_t)(gaddr >> 32) & 0x01FFFFFFu) | (2u << 30); // addr[56:32], type=2

    g1[0] = (int)((2u << 16) | (pad_en << 20) |
                  (pad_interval << 22) | (pad_amount << 25));     // data_size=4B + pad ctl
    g1[1] = (int)((tensor_d0 & 0xFFFFu) << 16);                   // tensor_dim0[15:0]
    g1[2] = (int)((tensor_d0 >> 16) | ((tensor_d1 & 0xFFFFu) << 16));
    g1[3] = (int)((tensor_d1 >> 16) | (tile_d0 << 16));           // tile_dim0
    g1[4] = (int)tile_d1;                                         // tile_dim1 (tile_dim2=0)
    g1[5] = (int)(uint32_t)stride0_elems;                         // tensor_dim0_stride lo
    g1[6] = (int)(uint32_t)((stride0_elems >> 32) & 0xFFFFu);     // stride hi; dim1_stride=0
    g1[7] = 0;
    __builtin_amdgcn_tensor_load_to_lds(g0, g1, g2, g3, g4, 0);
}

// ---------------------------------------------------------------------------
// Kernel 1: cost matrices. One workgroup = 16 rows x 1024 targets of one matrix.
// TDM async-stages the 16x152 logits tile into LDS while threads build the
// target table; WMMA f32 16x16x4 (ones-B) computes all 16 softmax row-sums
// on the matrix pipe. Hot loop uses v_rcp_f32 instead of IEEE div chains to
// stay at the HBM store roofline instead of going VALU-bound.
// ---------------------------------------------------------------------------
__global__ __launch_bounds__(256) void hm_cost_kernel(
    const float* __restrict__ sub_logits, const float* __restrict__ obj_logits,
    const float* __restrict__ sub_boxes,  const float* __restrict__ obj_boxes,
    const float* __restrict__ sub_tbox,   const float* __restrict__ obj_tbox,
    const int* __restrict__ sub_tids,     const int* __restrict__ obj_tids,
    float* __restrict__ outC)
{
    __shared__ float s_logits[TILE_R * NCd];   // 16x152 fp32 (TDM dest)
    __shared__ float s_tgt[NTd * 9];           // cx,cy,w,h,x0,y0,x1,y1,area (stride 9: bank-clean)
    __shared__ int   s_tid[NTd];
    __shared__ float s_row[TILE_R * 11];       // pred cx,cy,w,h,x0,y0,x1,y1,area,max,inv_sum

    const int tid  = threadIdx.x;
    const int lane = tid & 31;
    const int warp = tid >> 5;
    const int mat  = blockIdx.x / NTILES;      // 0 = sub, 1 = obj
    const int tile = blockIdx.x % NTILES;
    const int row0 = tile * TILE_R;

    const float* logits = mat ? obj_logits : sub_logits;
    const float* pbox   = mat ? obj_boxes  : sub_boxes;
    const float* tbox   = mat ? obj_tbox   : sub_tbox;
    const int*   tids   = mat ? obj_tids   : sub_tids;
    float* C = outC + (size_t)mat * CELEM;

    // Async: TDM pulls the logits tile while we populate the target table.
    if (warp == 0) {
        tdm_load_2d((uint32_t)(uintptr_t)s_logits,
                    (uint64_t)(uintptr_t)(logits + (size_t)row0 * NCd),
                    NCd, ROWS, NCd, TILE_R, NCd, 0, 0, 0);
    }

    for (int j = tid; j < NTd; j += 256) {
        float cx = tbox[j * 4 + 0], cy = tbox[j * 4 + 1];
        float w  = tbox[j * 4 + 2], h  = tbox[j * 4 + 3];
        float* t = &s_tgt[j * 9];
        t[0] = cx; t[1] = cy; t[2] = w; t[3] = h;
        t[4] = cx - 0.5f * w; t[5] = cy - 0.5f * h;
        t[6] = cx + 0.5f * w; t[7] = cy + 0.5f * h;
        t[8] = w * h;
        s_tid[j] = tids[j];
    }
    if (tid < TILE_R) {
        float cx = pbox[(size_t)(row0 + tid) * 4 + 0];
        float cy = pbox[(size_t)(row0 + tid) * 4 + 1];
        float w  = pbox[(size_t)(row0 + tid) * 4 + 2];
        float h  = pbox[(size_t)(row0 + tid) * 4 + 3];
        float* rp = &s_row[tid * 11];
        rp[0] = cx; rp[1] = cy; rp[2] = w; rp[3] = h;
        rp[4] = cx - 0.5f * w; rp[5] = cy - 0.5f * h;
        rp[6] = cx + 0.5f * w; rp[7] = cy + 0.5f * h;
        rp[8] = w * h;
    }
    if (warp == 0) __builtin_amdgcn_s_wait_tensorcnt(0);
    __syncthreads();

    // Row maxima: wave w handles rows 2w, 2w+1; lane-strided + xor-shuffle tree.
    for (int r = warp * 2; r < warp * 2 + 2; ++r) {
        float m = -3.4e38f;
        for (int k = lane; k < NCd; k += 32) m = fmaxf(m, s_logits[r * NCd + k]);
        for (int off = 16; off; off >>= 1) m = fmaxf(m, __shfl_xor(m, off, 32));
        if (lane == 0) s_row[r * 11 + 9] = m;
    }
    __syncthreads();

    // Softmax denominators for all 16 rows at once: D = exp(A) x ones, on the
    // matrix pipe. A-layout (f32 16x4): lanes 0-15 row M carry K={0,1},
    // lanes 16-31 carry K={2,3}. 152 = 38 * 4 exactly.
    if (warp == 0) {
        const int rowA = lane & 15;
        const int koff = (lane >> 4) * 2;
        const float mx = s_row[rowA * 11 + 9];
        v8f acc = {0.f, 0.f, 0.f, 0.f, 0.f, 0.f, 0.f, 0.f};
        v2f bones; bones.x = 1.0f; bones.y = 1.0f;
        for (int k = 0; k < NCd; k += 4) {
            v2f a;
            a.x = __expf(s_logits[rowA * NCd + k + koff]     - mx);
            a.y = __expf(s_logits[rowA * NCd + k + koff + 1] - mx);
            acc = __builtin_amdgcn_wmma_f32_16x16x4_f32(
                false, a, false, bones, (short)0, acc, false, false);
        }
        // C/D layout: VGPR r, lanes 0-15 -> M=r; lanes 16-31 -> M=8+r.
        if (lane == 0) {
            for (int r = 0; r < 8; ++r) s_row[r * 11 + 10] = __builtin_amdgcn_rcpf(acc[r]);
        } else if (lane == 16) {
            for (int r = 0; r < 8; ++r) s_row[(8 + r) * 11 + 10] = __builtin_amdgcn_rcpf(acc[r]);
        }
    }
    __syncthreads();

    // Main loop: per row, hoist pred-row state into registers; 4 col-chunks of
    // 256 (consecutive tids -> consecutive cols -> fully coalesced stores).
    for (int r = 0; r < TILE_R; ++r) {
        const float* rp = &s_row[r * 11];
        const float pcx = rp[0], pcy = rp[1], pw = rp[2], ph = rp[3];
        const float px0 = rp[4], py0 = rp[5], px1 = rp[6], py1 = rp[7];
        const float pa = rp[8], pmax = rp[9], pinv = rp[10];
        const float* lrow = &s_logits[r * NCd];
        float* Crow = C + (size_t)(row0 + r) * NTd;

        for (int cblk = 0; cblk < NTd / 256; ++cblk) {
            const int col = cblk * 256 + tid;
            const float* t = &s_tgt[col * 9];

            float l1 = fabsf(pcx - t[0]) + fabsf(pcy - t[1]) +
                       fabsf(pw  - t[2]) + fabsf(ph  - t[3]);

            float ix0 = fmaxf(px0, t[4]), iy0 = fmaxf(py0, t[5]);
            float ix1 = fminf(px1, t[6]), iy1 = fminf(py1, t[7]);
            float iw = fmaxf(ix1 - ix0, 0.f), ih = fmaxf(iy1 - iy0, 0.f);
            float inter = iw * ih;
            float uni   = pa + t[8] - inter;
            float cx0 = fminf(px0, t[4]), cy0 = fminf(py0, t[5]);
            float cx1 = fmaxf(px1, t[6]), cy1 = fmaxf(py1, t[7]);
            float cw = fmaxf(cx1 - cx0, 0.f), ch = fmaxf(cy1 - cy0, 0.f);
            float ca = cw * ch;
            // v_rcp_f32 (1 op) instead of ~10-op IEEE div chains
            float giou = inter * __builtin_amdgcn_rcpf(uni) -
                         (ca - uni) * __builtin_amdgcn_rcpf(ca);

            float cc = -__expf(lrow[s_tid[col]] - pmax) * pinv;

            Crow[col] = l1 + cc - giou;
        }
    }
}

// ---------------------------------------------------------------------------
// Kernel 2: exact JV Hungarian, one workgroup per (matrix, batch) = 64 blocks.
// TDM stages the strided 600x32 diagonal block (L2-resident) into LDS with
// hardware padding: +2 dwords every 32 dwords -> row stride 34 (conflict-free
// column reads since the 34-dword lane spread covers 32 distinct banks).
// ---------------------------------------------------------------------------
__global__ __launch_bounds__(256) void hm_lsa_kernel(float* __restrict__ dout)
{
    __shared__ float s_cost[Qd * 34];   // cost_T[q][t], t = target row, q = pred col
    __shared__ float s_v[Qd + 1];
    __shared__ float s_minv[Qd + 1];
    __shared__ int   s_way[Qd + 1];
    __shared__ int   s_used[Qd + 1];
    __shared__ float s_u[Td + 1];
    __shared__ int   s_p[Qd + 1];
    __shared__ float s_redv[8];
    __shared__ int   s_redj[8];
    __shared__ int   s_i0, s_j0, s_done;
    __shared__ float s_delta;

    const int tid  = threadIdx.x;
    const int lane = tid & 31;
    const int warp = tid >> 5;
    const int mat  = blockIdx.x >> 5;
    const int b    = blockIdx.x & 31;
    const float* C = dout + (size_t)mat * CELEM;

    if (warp == 0) {
        // tile: rows q in [b*600, b*600+600), cols [b*32, b*32+32) of [19200,1024]
        tdm_load_2d((uint32_t)(uintptr_t)s_cost,
                    (uint64_t)(uintptr_t)(C + (size_t)(b * Qd) * NTd + b * Td),
                    NTd, ROWS, Td, Qd, NTd,
                    /*pad_en=*/1, /*interval: 32 dwords=code 4*/4, /*amount: 2 dwords=code 1*/1);
    }
    for (int j = tid; j <= Qd; j += 256) { s_v[j] = 0.f; s_p[j] = 0; }
    if (tid <= Td) s_u[tid] = 0.f;
    if (warp == 0) __builtin_amdgcn_s_wait_tensorcnt(0);
    __syncthreads();

    for (int i = 1; i <= Td; ++i) {
        for (int j = tid; j <= Qd; j += 256) { s_minv[j] = 1e30f; s_used[j] = 0; s_way[j] = 0; }
        if (tid == 0) { s_p[0] = i; s_j0 = 0; s_done = 0; }
        __syncthreads();

        while (true) {
            if (tid == 0) { s_used[s_j0] = 1; s_i0 = s_p[s_j0]; }
            __syncthreads();
            const int   j0  = s_j0;
            const int   i0  = s_i0;
            const float ui0 = s_u[i0];

            float bv = 1e30f; int bj = 0;
            for (int j = tid + 1; j <= Qd; j += 256) {
                if (!s_used[j]) {
                    float cur = s_cost[(j - 1) * 34 + (i0 - 1)] - ui0 - s_v[j];
                    if (cur < s_minv[j]) { s_minv[j] = cur; s_way[j] = j0; }
                    float mv = s_minv[j];
                    if (mv < bv) { bv = mv; bj = j; }
                }
            }
            for (int off = 16; off; off >>= 1) {
                float ov = __shfl_xor(bv, off, 32);
                int   oj = __shfl_xor(bj, off, 32);
                if (ov < bv || (ov == bv && oj < bj)) { bv = ov; bj = oj; }
            }
            if (lane == 0) { s_redv[warp] = bv; s_redj[warp] = bj; }
            __syncthreads();
            if (tid == 0) {
                float dv = s_redv[0]; int dj = s_redj[0];
                for (int w = 1; w < 8; ++w)
                    if (s_redv[w] < dv || (s_redv[w] == dv && s_redj[w] < dj)) {
                        dv = s_redv[w]; dj = s_redj[w];
                    }
                s_delta = dv; s_j0 = dj; s_done = (s_p[dj] == 0);
            }
            __syncthreads();
            const float delta = s_delta;
            for (int j = tid; j <= Qd; j += 256) {
                if (s_used[j]) { s_v[j] -= delta; s_u[s_p[j]] += delta; } // p[j] distinct over used
                else             s_minv[j] -= delta;
            }
            __syncthreads();
            if (s_done) break;
        }
        if (tid == 0) {            // augment along 'way' chain
            int j0 = s_j0;
            while (j0 != 0) { int j1 = s_way[j0]; s_p[j0] = s_p[j1]; j0 = j1; }
        }
        __syncthreads();
    }

    if (tid == 0) {
        // cols sorted ascending (reference order); write indices as floats.
        float* ps = dout + 2 * CELEM + (size_t)mat * (2 * BSd * Td) + (size_t)b * Td;
        float* ts = ps + BSd * Td;
        int k = 0;
        for (int j = 1; j <= Qd; ++j)
            if (s_p[j] != 0) { ps[k] = (float)(j - 1); ts[k] = (float)(s_p[j] - 1); ++k; }
    }
}

extern "C" void kernel_launch(void* const* d_in, const int* in_sizes, int n_in,
                              void* d_out, int out_size, void* d_ws, size_t ws_size,
                              hipStream_t stream) {
    const float* sub_logits = (const float*)d_in[0];
    const float* obj_logits = (const float*)d_in[1];
    const float* sub_boxes  = (const float*)d_in[2];
    const float* obj_boxes  = (const float*)d_in[3];
    const float* sub_tbox   = (const float*)d_in[4];
    const float* obj_tbox   = (const float*)d_in[5];
    const int*   sub_tids   = (const int*)d_in[6];
    const int*   obj_tids   = (const int*)d_in[7];
    float* out = (float*)d_out;

    hm_cost_kernel<<<2 * NTILES, 256, 0, stream>>>(
        sub_logits, obj_logits, sub_boxes, obj_boxes,
        sub_tbox, obj_tbox, sub_tids, obj_tids, out);
    hm_lsa_kernel<<<2 * BSd, 256, 0, stream>>>(out);
}